// QNN_23527830848023
// MI455X (gfx1250) — compile-verified
//
#include <hip/hip_runtime.h>
#include <cstddef>

typedef float v2f __attribute__((ext_vector_type(2)));
typedef float v8f __attribute__((ext_vector_type(8)));

#define BN_EPS 1e-5f
#define PI_F 3.14159265358979323846f

struct QParams { const float* p[24]; };

// setup_inputs() dict order, params flattened recursively
enum {
  IX_X = 0, IX_W1, IX_B1, IX_BN1G, IX_BN1B, IX_BN1M, IX_BN1V,
  IX_W2, IX_B2, IX_QW,
  IX_W3, IX_B3, IX_BN2G, IX_BN2B, IX_BN2M, IX_BN2V,
  IX_W4, IX_B4, IX_BN3G, IX_BN3B, IX_BN3M, IX_BN3V,
  IX_W5, IX_B5
};

__device__ __forceinline__ v8f wmma4(v2f a, v2f b, v8f c) {
  // D(16x16,f32) = A(16x4,f32) * B(4x16,f32) + C ; exact fp32
  return __builtin_amdgcn_wmma_f32_16x16x4_f32(
      /*neg_a=*/false, a, /*neg_b=*/false, b,
      /*c_mod=*/(short)0, c, /*reuse_a=*/false, /*reuse_b=*/false);
}

// closed-form 1-qubit circuit: RY(angle); [RY(w),RZ(w),(RX(pi/2) if i<3)] x4 ; <Z>
__device__ __forceinline__ float expval_z(float angle, const float* qw) {
  float h = 0.5f * angle;
  float ar = __builtin_cosf(h), ai = 0.f;
  float br = __builtin_sinf(h), bi = 0.f;
  const float r = 0.70710678118654752f;
#pragma unroll
  for (int i = 0; i < 4; ++i) {
    float w = 0.5f * qw[i];
    float c = __builtin_cosf(w), s = __builtin_sinf(w);
    // RY(w)
    float t_ar = c * ar - s * br, t_ai = c * ai - s * bi;
    float t_br = s * ar + c * br, t_bi = s * ai + c * bi;
    // RZ(w): a *= (c,-s), b *= (c,+s)
    ar = t_ar * c + t_ai * s;  ai = t_ai * c - t_ar * s;
    br = t_br * c - t_bi * s;  bi = t_bi * c + t_br * s;
    if (i < 3) {
      // RX(pi/2): a' = r*(a - i b), b' = r*(-i a + b)
      float n_ar = r * (ar + bi), n_ai = r * (ai - br);
      float n_br = r * (br + ai), n_bi = r * (bi - ar);
      ar = n_ar; ai = n_ai; br = n_br; bi = n_bi;
    }
  }
  return (ar * ar + ai * ai) - (br * br + bi * bi);
}

__global__ __launch_bounds__(256) void qnn_wmma_kernel(QParams P,
                                                       float* __restrict__ out,
                                                       int tiles) {
  // per-wave h-tile buffer: 16 rows x stride 36 floats (16B-aligned rows, bank-safe)
  __shared__ __align__(16) float sm[8 * 576];
  // block-shared tiny params: [0..127]=pre_w2(32x4), [128..131]=pre_b2, [132..147]=q_weights(4x4)
  __shared__ __align__(16) float sp[160];

  const int tid  = threadIdx.x;
  const int wave = tid >> 5;
  const int lane = tid & 31;
  const int half = lane >> 4;   // lane-half selects K within A/B layouts
  const int nIdx = lane & 15;   // column (N) / batch-row (M) index

  if (tid < 128) sp[tid] = P.p[IX_W2][tid];
  if (tid < 4)   sp[128 + tid] = P.p[IX_B2][tid];
  if (tid < 16)  sp[132 + tid] = P.p[IX_QW][tid];
  __syncthreads();

  float* hbuf = sm + wave * 576;

  // ---- preload weights into WMMA B-layout registers: row k = v + 2*half (+4j) ----
  v2f w1B[2][4], w3B[2], w4B[8], w5B[4];
  const float* W1 = P.p[IX_W1];   // [16,32]
#pragma unroll
  for (int t = 0; t < 2; ++t)
#pragma unroll
    for (int j = 0; j < 4; ++j) {
      w1B[t][j].x = W1[(4 * j + 2 * half + 0) * 32 + 16 * t + nIdx];
      w1B[t][j].y = W1[(4 * j + 2 * half + 1) * 32 + 16 * t + nIdx];
    }
  const float* W3 = P.p[IX_W3];   // [4,32]
#pragma unroll
  for (int t = 0; t < 2; ++t) {
    w3B[t].x = W3[(2 * half + 0) * 32 + 16 * t + nIdx];
    w3B[t].y = W3[(2 * half + 1) * 32 + 16 * t + nIdx];
  }
  const float* W4 = P.p[IX_W4];   // [32,16]
#pragma unroll
  for (int j = 0; j < 8; ++j) {
    w4B[j].x = W4[(4 * j + 2 * half + 0) * 16 + nIdx];
    w4B[j].y = W4[(4 * j + 2 * half + 1) * 16 + nIdx];
  }
  const float* W5 = P.p[IX_W5];   // [16,10] -> pad N to 16 with zeros
  const bool colOK = (nIdx < 10);
#pragma unroll
  for (int j = 0; j < 4; ++j) {
    w5B[j].x = colOK ? W5[(4 * j + 2 * half + 0) * 10 + nIdx] : 0.f;
    w5B[j].y = colOK ? W5[(4 * j + 2 * half + 1) * 10 + nIdx] : 0.f;
  }

  // ---- per-column fused bias/BN constants (column == nIdx within tile) ----
  float b1c[2], sc1[2], sh1[2], b3c[2], sc2[2], sh2[2];
#pragma unroll
  for (int t = 0; t < 2; ++t) {
    int c = 16 * t + nIdx;
    b1c[t] = P.p[IX_B1][c];
    float s1 = P.p[IX_BN1G][c] * rsqrtf(P.p[IX_BN1V][c] + BN_EPS);
    sc1[t] = s1; sh1[t] = P.p[IX_BN1B][c] - P.p[IX_BN1M][c] * s1;
    b3c[t] = P.p[IX_B3][c];
    float s2 = P.p[IX_BN2G][c] * rsqrtf(P.p[IX_BN2V][c] + BN_EPS);
    sc2[t] = s2; sh2[t] = P.p[IX_BN2B][c] - P.p[IX_BN2M][c] * s2;
  }
  float b4c = P.p[IX_B4][nIdx];
  float s3  = P.p[IX_BN3G][nIdx] * rsqrtf(P.p[IX_BN3V][nIdx] + BN_EPS);
  float sc3 = s3, sh3 = P.p[IX_BN3B][nIdx] - P.p[IX_BN3M][nIdx] * s3;
  float b5c = colOK ? P.p[IX_B5][nIdx] : 0.f;

  // quantum-stage per-lane constants: this lane owns qubits q0=2*half, q0+1 for row nIdx
  const int q0 = 2 * half;
  const float pb2_0 = sp[128 + q0], pb2_1 = sp[128 + q0 + 1];
  float qw0[4], qw1[4];
#pragma unroll
  for (int i = 0; i < 4; ++i) {
    qw0[i] = sp[132 + q0 * 4 + i];
    qw1[i] = sp[132 + (q0 + 1) * 4 + i];
  }

  const float* X = P.p[IX_X];
  const v8f dz = {0.f, 0.f, 0.f, 0.f, 0.f, 0.f, 0.f, 0.f};
  const int totalWaves = gridDim.x * 8;

  for (int tile = blockIdx.x * 8 + wave; tile < tiles; tile += totalWaves) {
    const int row0 = tile << 4;

    // ===== layer 1: h1 = BN1(relu(X[16,16] @ W1[16,32] + b1)) via 8 WMMAs =====
    v2f a1[4];
    const float* xrow = X + (size_t)(row0 + nIdx) * 16 + 2 * half;
#pragma unroll
    for (int j = 0; j < 4; ++j) a1[j] = *(const v2f*)(xrow + 4 * j);

#pragma unroll
    for (int t = 0; t < 2; ++t) {
      v8f d = dz;
#pragma unroll
      for (int j = 0; j < 4; ++j) d = wmma4(a1[j], w1B[t][j], d);
#pragma unroll
      for (int r = 0; r < 8; ++r) {
        float z = d[r] + b1c[t];
        z = fmaxf(z, 0.f) * sc1[t] + sh1[t];
        hbuf[(r + 8 * half) * 36 + 16 * t + nIdx] = z;   // D -> row-major LDS
      }
    }

    // ===== layer 2 + quantum (VALU): lane computes (row=nIdx, qubits q0,q0+1) =====
    float hr[32];
    const float* hrow = hbuf + nIdx * 36;
#pragma unroll
    for (int kk = 0; kk < 8; ++kk)
      *(float4*)(hr + 4 * kk) = *(const float4*)(hrow + 4 * kk);

    float acc0 = pb2_0, acc1 = pb2_1;
#pragma unroll
    for (int k = 0; k < 32; ++k) {
      v2f wv = *(const v2f*)(sp + 4 * k + q0);
      acc0 = fmaf(hr[k], wv.x, acc0);
      acc1 = fmaf(hr[k], wv.y, acc1);
    }
    float q0v = expval_z(PI_F * tanhf(acc0), qw0);
    float q1v = expval_z(PI_F * tanhf(acc1), qw1);

    // ===== layer 3: h3 = BN2(relu(q[16,4] @ W3[4,32] + b3)); A is already lane-resident =====
    v2f aq; aq.x = q0v; aq.y = q1v;   // A(16x4): M=nIdx, K=2*half+{0,1}
#pragma unroll
    for (int t = 0; t < 2; ++t) {
      v8f d = wmma4(aq, w3B[t], dz);
#pragma unroll
      for (int r = 0; r < 8; ++r) {
        float z = d[r] + b3c[t];
        z = fmaxf(z, 0.f) * sc2[t] + sh2[t];
        hbuf[(r + 8 * half) * 36 + 16 * t + nIdx] = z;
      }
    }

    // ===== layer 4: h4 = BN3(relu(h3[16,32] @ W4[32,16] + b4)) via 8 WMMAs =====
    v8f d4 = dz;
#pragma unroll
    for (int j = 0; j < 8; ++j) {
      v2f a = *(const v2f*)(hbuf + nIdx * 36 + 4 * j + 2 * half);
      d4 = wmma4(a, w4B[j], d4);
    }
#pragma unroll
    for (int r = 0; r < 8; ++r) {
      float z = d4[r] + b4c;
      z = fmaxf(z, 0.f) * sc3 + sh3;
      hbuf[(r + 8 * half) * 36 + nIdx] = z;
    }

    // ===== layer 5: out = h4[16,16] @ W5[16,10] + b5 via 4 WMMAs (N padded) =====
    v8f d5 = dz;
#pragma unroll
    for (int j = 0; j < 4; ++j) {
      v2f a = *(const v2f*)(hbuf + nIdx * 36 + 4 * j + 2 * half);
      d5 = wmma4(a, w5B[j], d5);
    }
    if (colOK) {
#pragma unroll
      for (int r = 0; r < 8; ++r)
        out[(size_t)(row0 + r + 8 * half) * 10 + nIdx] = d5[r] + b5c;
    }
  }
}

extern "C" void kernel_launch(void* const* d_in, const int* in_sizes, int n_in,
                              void* d_out, int out_size, void* d_ws, size_t ws_size,
                              hipStream_t stream) {
  (void)d_ws; (void)ws_size; (void)out_size;
  QParams P;
  for (int i = 0; i < 24; ++i) P.p[i] = (i < n_in) ? (const float*)d_in[i] : nullptr;
  const int rows  = in_sizes[0] / 16;   // x is [B,16]
  const int tiles = rows / 16;          // 16 batch rows per wave
  dim3 grid(2048), block(256);          // 8 waves/block, grid-stride over tiles
  qnn_wmma_kernel<<<grid, block, 0, stream>>>(P, (float*)d_out, tiles);
}